// AttentionAggregation_63806034149757
// MI455X (gfx1250) — compile-verified
//
#include <hip/hip_runtime.h>
#include <math.h>

// AttentionAggregation for MI455X (gfx1250, wave32).
// 3-pass segment softmax + scatter, logits computed via V_WMMA_F32_16X16X4_F32.

typedef __attribute__((ext_vector_type(2))) float v2f;
typedef __attribute__((ext_vector_type(8))) float v8f;

#define HEADS 8
#define DHEAD 8
#define INV_SQRT_DHEAD 0.3535533905932738f

// ---------- atomic helpers (native instructions, no CAS loops) ----------

// float atomic max via monotonic int/uint punning (valid when slot init = -inf).
__device__ __forceinline__ void atomicMaxF32(float* addr, float val) {
  if (val >= 0.0f) {
    atomicMax((int*)addr, __float_as_int(val));           // global_atomic_max_i32
  } else {
    atomicMin((unsigned int*)addr, __float_as_uint(val)); // global_atomic_min_u32
  }
}

__device__ __forceinline__ void atomicAddF32(float* addr, float val) {
  unsafeAtomicAdd(addr, val); // global_atomic_add_f32 (no return)
}

// ---------- scalar logit (tail tiles / no-wcache fallback) ----------
__device__ __forceinline__ float computeLogit(const float* __restrict__ q,
                                              const float* __restrict__ k,
                                              const float* __restrict__ cut,
                                              long long row) {
  const float* qr = q + row * DHEAD;
  const float* kr = k + row * DHEAD;
  float s = 0.0f;
#pragma unroll
  for (int i = 0; i < DHEAD; ++i) s += qr[i] * kr[i];
  return s * cut[row >> 3] * INV_SQRT_DHEAD;
}

// ---------- pass 0: init out / m / denom ----------
__global__ void init_kernel(float* __restrict__ out, int outN,
                            float* __restrict__ m, float* __restrict__ denom, int nh) {
  int i = blockIdx.x * blockDim.x + threadIdx.x;
  if (i < outN) out[i] = 0.0f;
  if (i < nh) {
    m[i] = -INFINITY;
    denom[i] = 0.0f;
  }
}

// ---------- pass 1: logits via WMMA + segment max ----------
// One wave handles a tile of 16 (edge,head) rows (= 2 edges).
// A = q-tile [16 x 4], B = k-tile^T [4 x 16]; diag(A*B) = per-row dot products.
// 32-bit A layout: VGPR0 lanes0-15 K=0 / lanes16-31 K=2; VGPR1 K=1/K=3.
// B (= K^T) uses the mirrored layout, so per-lane addressing is identical.
//
// D layout: VGPR r holds diag(r) in lane r and diag(8+r) in lane 24+r, so
// lanes 0-7 and 24-31 already own one diagonal element locally. We only need
// a 3-level cndmask tree to pick acc[lane&7] (no cross-lane traffic at all).
__global__ void logits_kernel(const float* __restrict__ q,
                              const float* __restrict__ k,
                              const float* __restrict__ cut,
                              const int* __restrict__ dst,
                              float* __restrict__ m,
                              float* __restrict__ wc,   // may be nullptr
                              long long totalRows) {
  const int lane = threadIdx.x & 31;
  const int waveId = threadIdx.x >> 5;
  const long long rowBase = ((long long)blockIdx.x * 8 + waveId) * 16;
  if (rowBase >= totalRows) return;

  float wlog = 0.0f;
  long long row;
  bool active;

  if (rowBase + 16 <= totalRows) {
    // Full tile: EXEC is all ones -> WMMA legal.
    const int r = lane & 15;
    const int hi = lane >> 4;
    const float* qr = q + (rowBase + r) * DHEAD + 2 * hi;
    const float* kr = k + (rowBase + r) * DHEAD + 2 * hi;

    v2f a01; a01.x = qr[0]; a01.y = qr[1];   // K = {0,1} or {2,3}
    v2f a45; a45.x = qr[4]; a45.y = qr[5];   // K = {4,5} or {6,7}
    v2f b01; b01.x = kr[0]; b01.y = kr[1];
    v2f b45; b45.x = kr[4]; b45.y = kr[5];

    v8f acc = {};
    acc = __builtin_amdgcn_wmma_f32_16x16x4_f32(false, a01, false, b01,
                                                (short)0, acc, false, false);
    acc = __builtin_amdgcn_wmma_f32_16x16x4_f32(false, a45, false, b45,
                                                (short)0, acc, false, false);

    // Register-index select: wlog = acc[lane & 7] via cndmask binary tree.
    const int sel = lane & 7;
    float t01 = (sel & 1) ? acc[1] : acc[0];
    float t23 = (sel & 1) ? acc[3] : acc[2];
    float t45 = (sel & 1) ? acc[5] : acc[4];
    float t67 = (sel & 1) ? acc[7] : acc[6];
    float t0123 = (sel & 2) ? t23 : t01;
    float t4567 = (sel & 2) ? t67 : t45;
    wlog = (sel & 4) ? t4567 : t0123;

    // lane rr (rr<8) owns diag(rr) -> row rowBase+rr
    // lane 24+rr     owns diag(8+rr) -> row rowBase+8+rr
    active = (lane < 8) || (lane >= 24);
    row = rowBase + ((lane < 8) ? lane : (lane - 16));
  } else {
    // Tail tile: scalar dot per active row (lanes 0..15).
    row = rowBase + lane;
    active = (lane < 16) && (row < totalRows);
    if (active) {
      const float* qr = q + row * DHEAD;
      const float* kr = k + row * DHEAD;
      float s = 0.0f;
#pragma unroll
      for (int i = 0; i < DHEAD; ++i) s += qr[i] * kr[i];
      wlog = s;
    }
  }

  if (active) {
    long long e = row >> 3;
    int h = (int)(row & 7);
    float wfin = wlog * cut[e] * INV_SQRT_DHEAD;
    if (wc) wc[row] = wfin;
    atomicMaxF32(&m[(long long)dst[e] * HEADS + h], wfin);
  }
}

// ---------- pass 2: denom = segment_sum(exp(w - m[dst])) ----------
__global__ void denom_kernel(const float* __restrict__ q,
                             const float* __restrict__ k,
                             const float* __restrict__ cut,
                             const int* __restrict__ dst,
                             const float* __restrict__ m,
                             float* __restrict__ denom,
                             const float* __restrict__ wc,
                             long long totalRows) {
  long long i = (long long)blockIdx.x * blockDim.x + threadIdx.x;
  if (i >= totalRows) return;
  long long e = i >> 3;
  int h = (int)(i & 7);
  float wfin = wc ? wc[i] : computeLogit(q, k, cut, i);
  long long o = (long long)dst[e] * HEADS + h;
  float ex = __expf(wfin - m[o]);
  atomicAddF32(&denom[o], ex);
}

// ---------- pass 3: out[dst] += alpha * v ----------
__global__ void scatter_kernel(const float* __restrict__ q,
                               const float* __restrict__ k,
                               const float* __restrict__ v,
                               const float* __restrict__ cut,
                               const int* __restrict__ dst,
                               const float* __restrict__ m,
                               const float* __restrict__ denom,
                               const float* __restrict__ wc,
                               float* __restrict__ out,
                               long long totalRows) {
  long long i = (long long)blockIdx.x * blockDim.x + threadIdx.x;
  if (i >= totalRows) return;
  long long e = i >> 3;
  int h = (int)(i & 7);
  float wfin = wc ? wc[i] : computeLogit(q, k, cut, i);
  long long o = (long long)dst[e] * HEADS + h;
  float alpha = __expf(wfin - m[o]) / denom[o];

  // v[e, h*8 + j] == v[(e*8+h)*8 + j]; force two b128 loads.
  const float4* vp = (const float4*)(v + i * DHEAD);
  float4 v0 = vp[0];
  float4 v1 = vp[1];
  float* op = out + o * DHEAD;       // out[d*64 + h*8 + j] == out[(d*8+h)*8 + j]
  atomicAddF32(&op[0], alpha * v0.x);
  atomicAddF32(&op[1], alpha * v0.y);
  atomicAddF32(&op[2], alpha * v0.z);
  atomicAddF32(&op[3], alpha * v0.w);
  atomicAddF32(&op[4], alpha * v1.x);
  atomicAddF32(&op[5], alpha * v1.y);
  atomicAddF32(&op[6], alpha * v1.z);
  atomicAddF32(&op[7], alpha * v1.w);
}

// ---------- launcher ----------
extern "C" void kernel_launch(void* const* d_in, const int* in_sizes, int n_in,
                              void* d_out, int out_size, void* d_ws, size_t ws_size,
                              hipStream_t stream) {
  const float* key   = (const float*)d_in[0];
  const float* value = (const float*)d_in[1];
  const float* query = (const float*)d_in[2];
  const float* cut   = (const float*)d_in[3];
  const int*   eidx  = (const int*)d_in[4];

  const long long E = (long long)in_sizes[0] / 64;   // edges
  const int N = out_size / 64;                       // nodes
  const int* dst = eidx + E;                         // edge_index[1]
  const long long totalRows = E * HEADS;

  // workspace layout: m[N*8] | denom[N*8] | (optional) wcache[E*8]
  float* m     = (float*)d_ws;
  float* denom = m + (size_t)N * HEADS;
  size_t baseBytes = (size_t)2 * N * HEADS * sizeof(float);
  float* wc = nullptr;
  if (ws_size >= baseBytes + (size_t)totalRows * sizeof(float)) {
    wc = denom + (size_t)N * HEADS;
  }

  float* out = (float*)d_out;

  // pass 0: init
  {
    int nh = N * HEADS;
    int maxN = out_size > nh ? out_size : nh;
    int blocks = (maxN + 255) / 256;
    init_kernel<<<blocks, 256, 0, stream>>>(out, out_size, m, denom, nh);
  }
  // pass 1: logits (WMMA) + segment max. 8 waves/block * 16 rows = 128 rows/block.
  {
    long long blocks = (totalRows + 127) / 128;
    logits_kernel<<<(int)blocks, 256, 0, stream>>>(query, key, cut, dst, m, wc,
                                                   totalRows);
  }
  // pass 2: denom
  {
    long long blocks = (totalRows + 255) / 256;
    denom_kernel<<<(int)blocks, 256, 0, stream>>>(query, key, cut, dst, m, denom,
                                                  wc, totalRows);
  }
  // pass 3: scatter weighted values
  {
    long long blocks = (totalRows + 255) / 256;
    scatter_kernel<<<(int)blocks, 256, 0, stream>>>(query, key, value, cut, dst,
                                                    m, denom, wc, out, totalRows);
  }
}